// ContextEncoder_76287209112005
// MI455X (gfx1250) — compile-verified
//
#include <hip/hip_runtime.h>
#include <hip/hip_bf16.h>

typedef float v2f __attribute__((ext_vector_type(2)));
typedef float v8f __attribute__((ext_vector_type(8)));
typedef int   v4i __attribute__((ext_vector_type(4)));

#define B_BATCH 4
#define T_SEQ   1024
#define D_MODEL 1024
#define NH      16
#define HD      64
#define QKV_LD  3072
#define MAXS    1024
#define QT_CNT  (T_SEQ / 16)
#define KC      32            // K-chunk staged in LDS per buffer

#if defined(__has_builtin)
#if __has_builtin(__builtin_amdgcn_global_load_async_to_lds_b128)
#define HAVE_ASYNC_LDS 1
#endif
#endif

typedef __attribute__((address_space(1))) v4i* gptr4_t;
typedef __attribute__((address_space(3))) v4i* lptr4_t;

// ---------------------------------------------------------------------------
// K1: theta = x * mask   (float4 vectorized; theta lives in d_out)
// ---------------------------------------------------------------------------
__global__ void __launch_bounds__(256) ce_mask_theta(
    const float4* __restrict__ x, const int* __restrict__ lengths,
    float4* __restrict__ theta) {
  int i  = blockIdx.x * blockDim.x + threadIdx.x;   // over B*T*D/4
  int d4 = D_MODEL / 4;
  int t  = (i / d4) % T_SEQ;
  int b  = i / (d4 * T_SEQ);
  float4 v = x[i];
  if (t >= lengths[b]) { v.x = 0.f; v.y = 0.f; v.z = 0.f; v.w = 0.f; }
  theta[i] = v;
}

// ---------------------------------------------------------------------------
// K2: qkv[4096,3072] = theta[4096,1024] @ W[1024,3072] + b
// Block (4 waves) computes a 128x64 tile; the shared 32x64 W tile is staged
// into double-buffered LDS with global_load_async_to_lds_b128 (ASYNCcnt),
// then fragments are fed from LDS (rows k and k+1 are 256B apart ->
// ds_load_2addr_b32 merge). Each wave owns a 32x64 sub-tile.
// A 16x4 layout: lane L holds A[L][k0+co], A[L][k0+co+1], co = (lane<16)?0:2
// B 4x16 layout: VGPR0 = rows k0+co striped over lanes; VGPR1 = rows k0+co+1
// ---------------------------------------------------------------------------
__global__ void __launch_bounds__(128) ce_qkv_gemm(
    const float* __restrict__ theta, const float* __restrict__ w,
    const float* __restrict__ bias, float* __restrict__ qkv) {
  __shared__ float wtile[2][KC * 64];     // 2 x 8KB

  int tid  = threadIdx.x;
  int wv   = tid >> 5;
  int lane = tid & 31;
  int l16  = lane & 15;
  bool lo  = lane < 16;
  int co   = lo ? 0 : 2;

  const int nTiles = QKV_LD / 64;               // 48
  int n0 = (blockIdx.x % nTiles) * 64;
  int m0 = (blockIdx.x / nTiles) * 128 + wv * 32;

  v8f acc[2][4];
#pragma unroll
  for (int nt = 0; nt < 4; ++nt) {
    float bb = bias[n0 + nt * 16 + l16];
    v8f c;
#pragma unroll
    for (int r = 0; r < 8; ++r) c[r] = bb;
    acc[0][nt] = c; acc[1][nt] = c;
  }

  const float* a0 = theta + (size_t)(m0 + l16) * D_MODEL + co;
  const float* a1 = a0 + 16 * D_MODEL;

  // Stage W[k0..k0+KC) x [n0..n0+64) into wtile[buf]: 512 16B chunks / 128 thr
  auto stage = [&](int buf, int k0) {
#pragma unroll
    for (int c = 0; c < 4; ++c) {
      int chunk = tid + 128 * c;        // 0..511
      int r  = chunk >> 4;              // row within chunk    0..31
      int cc = (chunk & 15) << 2;       // col (floats)        0..60
      const float* g = w + (size_t)(k0 + r) * QKV_LD + n0 + cc;
      float* l = &wtile[buf][r * 64 + cc];
#if defined(HAVE_ASYNC_LDS)
      __builtin_amdgcn_global_load_async_to_lds_b128(
          (gptr4_t)(size_t)g, (lptr4_t)(unsigned)(size_t)l, 0, 0);
#else
      *(float4*)l = *(const float4*)g;
#endif
    }
  };

  stage(0, 0);
  for (int kc = 0; kc < D_MODEL / KC; ++kc) {   // 32 chunks
    int buf = kc & 1;
#if defined(HAVE_ASYNC_LDS)
    asm volatile("s_wait_asynccnt 0x0" ::: "memory");
#endif
    __syncthreads();
    if (kc + 1 < D_MODEL / KC) stage(buf ^ 1, (kc + 1) * KC);

    const float* wt = &wtile[buf][0];
    int k0g = kc * KC;
#pragma unroll
    for (int ks = 0; ks < KC / 4; ++ks) {
      v2f af0 = *(const v2f*)(a0 + k0g + 4 * ks);
      v2f af1 = *(const v2f*)(a1 + k0g + 4 * ks);
      const float* wr0 = wt + (4 * ks + co) * 64 + l16;
      const float* wr1 = wr0 + 64;
#pragma unroll
      for (int nt = 0; nt < 4; ++nt) {
        v2f bf; bf.x = wr0[nt * 16]; bf.y = wr1[nt * 16];
        acc[0][nt] = __builtin_amdgcn_wmma_f32_16x16x4_f32(
            false, af0, false, bf, (short)0, acc[0][nt], false, false);
        acc[1][nt] = __builtin_amdgcn_wmma_f32_16x16x4_f32(
            false, af1, false, bf, (short)0, acc[1][nt], false, false);
      }
    }
  }

#pragma unroll
  for (int hh = 0; hh < 2; ++hh)
#pragma unroll
    for (int nt = 0; nt < 4; ++nt)
#pragma unroll
      for (int r = 0; r < 8; ++r) {
        int row = m0 + hh * 16 + r + (lo ? 0 : 8);
        qkv[(size_t)row * QKV_LD + n0 + nt * 16 + l16] = acc[hh][nt][r];
      }
}

// ---------------------------------------------------------------------------
// K3: flash attention + masked pooling partials.
// One wave per (b, h, 16-query tile). Computes S^T = K @ Q^T so softmax rows
// map to lanes (one shfl_xor(16) per reduction). Emits pooled partial sums
// per tile (deterministic -- no float atomics).
// ---------------------------------------------------------------------------
__global__ void __launch_bounds__(128) ce_attn_pool(
    const float* __restrict__ qkv, const int* __restrict__ lengths,
    const float* __restrict__ rel_pos, float* __restrict__ partial) {
  int wave = (blockIdx.x * blockDim.x + threadIdx.x) >> 5;
  int lane = threadIdx.x & 31;
  int q16  = lane & 15;
  bool lo  = lane < 16;
  int co   = lo ? 0 : 2;

  int qt = wave % QT_CNT;
  int h  = (wave / QT_CNT) % NH;
  int b  = wave / (QT_CNT * NH);
  int len = lengths[b];
  int tq0 = qt * 16;

  float* pp = partial + ((size_t)((b * NH + h) * QT_CNT + qt)) * HD + q16;

  if (tq0 >= len) {                 // pooled contribution is exactly zero
    if (lo) {
#pragma unroll
      for (int nt = 0; nt < 4; ++nt) pp[nt * 16] = 0.f;
    }
    return;
  }

  int tq = tq0 + q16;

  // Q fragments (identical layout whether used as A or as B=Q^T):
  const float* qb = qkv + (size_t)(b * T_SEQ + tq0 + q16) * QKV_LD + h * HD + co;
  v2f qf[16];
#pragma unroll
  for (int j = 0; j < 16; ++j) qf[j] = *(const v2f*)(qb + 4 * j);

  float m = -3.0e38f, l = 0.f;
  v8f acc[4];
#pragma unroll
  for (int nt = 0; nt < 4; ++nt) {
    v8f z;
#pragma unroll
    for (int r = 0; r < 8; ++r) z[r] = 0.f;
    acc[nt] = z;
  }

  int nkt = (len + 15) >> 4;
  for (int kt = 0; kt < nkt; ++kt) {
    int tk0 = kt * 16;

    // S^T = K(16x64) @ Q^T : A = K tile, B = Q fragments
    const float* kb = qkv + (size_t)(b * T_SEQ + tk0 + q16) * QKV_LD +
                      D_MODEL + h * HD + co;
    v8f s = {0.f, 0.f, 0.f, 0.f, 0.f, 0.f, 0.f, 0.f};
#pragma unroll
    for (int j = 0; j < 16; ++j) {
      v2f kf = *(const v2f*)(kb + 4 * j);
      s = __builtin_amdgcn_wmma_f32_16x16x4_f32(
          false, kf, false, qf[j], (short)0, s, false, false);
    }

    // s[r] = score(key = tk0 + r + (lo?0:8), query = tq): scale + bias + mask
    int keyb = tk0 + (lo ? 0 : 8);
    float p[8];
    float mt = -3.0e38f;
#pragma unroll
    for (int r = 0; r < 8; ++r) {
      int tk = keyb + r;
      float sc = s[r] * 0.125f + rel_pos[(tq - tk + (MAXS - 1)) * NH + h];
      sc = (tk < len) ? sc : -1.0e30f;
      p[r] = sc;
      mt = fmaxf(mt, sc);
    }
    mt = fmaxf(mt, __shfl_xor(mt, 16, 32));   // full 16-key max per query lane
    float mnew = fmaxf(m, mt);
    float scl  = __expf(m - mnew);
    float ls = 0.f;
#pragma unroll
    for (int r = 0; r < 8; ++r) { p[r] = __expf(p[r] - mnew); ls += p[r]; }
    ls += __shfl_xor(ls, 16, 32);
    l = l * scl + ls;
    m = mnew;

    // rescale accumulators: row r maps to query r (lo) / r+8 (hi)
#pragma unroll
    for (int r = 0; r < 8; ++r) {
      float sr = __shfl(scl, lo ? r : r + 8, 32);
      acc[0][r] *= sr; acc[1][r] *= sr; acc[2][r] *= sr; acc[3][r] *= sr;
    }

    // P^T -> A fragments via half-swaps, then acc += P @ V
    float xs[8];
#pragma unroll
    for (int r = 0; r < 8; ++r) xs[r] = __shfl_xor(p[r], 16, 32);

    const float* vb = qkv + (size_t)(b * T_SEQ + tk0) * QKV_LD +
                      2 * D_MODEL + h * HD + q16;
#pragma unroll
    for (int j = 0; j < 4; ++j) {
      int bs = (j & 1) * 4;
      v2f av;
      if (j < 2) { av.x = lo ? p[bs]     : xs[bs + 2];
                   av.y = lo ? p[bs + 1] : xs[bs + 3]; }
      else       { av.x = lo ? xs[bs]     : p[bs + 2];
                   av.y = lo ? xs[bs + 1] : p[bs + 3]; }
      const float* vr0 = vb + (size_t)(4 * j + co) * QKV_LD;
      const float* vr1 = vr0 + QKV_LD;
#pragma unroll
      for (int nt = 0; nt < 4; ++nt) {
        v2f bv; bv.x = vr0[nt * 16]; bv.y = vr1[nt * 16];
        acc[nt] = __builtin_amdgcn_wmma_f32_16x16x4_f32(
            false, av, false, bv, (short)0, acc[nt], false, false);
      }
    }
  }

  // pooled partial: sum over valid queries of acc/l
  float contrib = (tq < len) ? (1.0f / l) : 0.0f;
  float cb[8];
#pragma unroll
  for (int r = 0; r < 8; ++r) cb[r] = __shfl(contrib, lo ? r : r + 8, 32);

#pragma unroll
  for (int nt = 0; nt < 4; ++nt) {
    float cs = 0.f;
#pragma unroll
    for (int r = 0; r < 8; ++r) cs += acc[nt][r] * cb[r];
    cs += __shfl_xor(cs, 16, 32);
    if (lo) pp[nt * 16] = cs;
  }
}

// ---------------------------------------------------------------------------
// K4: reduce partials over query tiles, /denom, LayerNorm, clip. 1 block / b.
// ---------------------------------------------------------------------------
__global__ void __launch_bounds__(256) ce_pool_ln(
    const float* __restrict__ partial, const int* __restrict__ lengths,
    float* __restrict__ out) {
  int b = blockIdx.x, tid = threadIdx.x;
  __shared__ float sred[8];
  int len = lengths[b];
  float denom = (float)(len < 1 ? 1 : len);

  float v[4];
#pragma unroll
  for (int i = 0; i < 4; ++i) {
    int d = tid + 256 * i;
    int h = d >> 6, hc = d & 63;
    const float* p = partial + (size_t)(b * NH + h) * QT_CNT * HD + hc;
    float s = 0.f;
    for (int q = 0; q < QT_CNT; ++q) s += p[(size_t)q * HD];
    v[i] = s / denom;
  }

  float s = v[0] + v[1] + v[2] + v[3];
#pragma unroll
  for (int o = 16; o > 0; o >>= 1) s += __shfl_xor(s, o, 32);
  if ((tid & 31) == 0) sred[tid >> 5] = s;
  __syncthreads();
  float tot = 0.f;
#pragma unroll
  for (int i = 0; i < 8; ++i) tot += sred[i];
  float mean = tot * (1.0f / (float)D_MODEL);
  __syncthreads();

  float sv = 0.f;
#pragma unroll
  for (int i = 0; i < 4; ++i) { float dd = v[i] - mean; sv += dd * dd; }
#pragma unroll
  for (int o = 16; o > 0; o >>= 1) sv += __shfl_xor(sv, o, 32);
  if ((tid & 31) == 0) sred[tid >> 5] = sv;
  __syncthreads();
  float tv = 0.f;
#pragma unroll
  for (int i = 0; i < 8; ++i) tv += sred[i];
  float rinv = rsqrtf(tv * (1.0f / (float)D_MODEL) + 1e-5f);

#pragma unroll
  for (int i = 0; i < 4; ++i) {
    float o = (v[i] - mean) * rinv;
    o = fminf(10.0f, fmaxf(-10.0f, o));
    out[b * D_MODEL + tid + 256 * i] = o;
  }
}

// ---------------------------------------------------------------------------
extern "C" void kernel_launch(void* const* d_in, const int* in_sizes, int n_in,
                              void* d_out, int out_size, void* d_ws, size_t ws_size,
                              hipStream_t stream) {
  (void)in_sizes; (void)n_in; (void)out_size; (void)ws_size;
  const float* x       = (const float*)d_in[0];
  const int*   lengths = (const int*)d_in[1];
  const float* w_qkv   = (const float*)d_in[2];
  const float* b_qkv   = (const float*)d_in[3];
  const float* rel_pos = (const float*)d_in[4];

  float* theta   = (float*)d_out;                               // [B,T,D]
  float* context = theta + (size_t)B_BATCH * T_SEQ * D_MODEL;   // [B,D]

  float* qkv     = (float*)d_ws;                                // [4096,3072]
  float* partial = qkv + (size_t)B_BATCH * T_SEQ * QKV_LD;      // [B,H,QT,HD]

  // K1: theta = x * mask
  {
    int n4 = B_BATCH * T_SEQ * D_MODEL / 4;
    ce_mask_theta<<<n4 / 256, 256, 0, stream>>>((const float4*)x, lengths,
                                                (float4*)theta);
  }
  // K2: qkv GEMM (128x64 per block, async-LDS staged W, 4 waves/block)
  {
    int blocks = (B_BATCH * T_SEQ / 128) * (QKV_LD / 64);   // 32*48 = 1536
    ce_qkv_gemm<<<blocks, 128, 0, stream>>>(theta, w_qkv, b_qkv, qkv);
  }
  // K3: attention + pooled partials (1 wave per 16-query tile)
  {
    int waves = B_BATCH * NH * QT_CNT;                      // 4096
    ce_attn_pool<<<waves / 4, 128, 0, stream>>>(qkv, lengths, rel_pos, partial);
  }
  // K4: pool + layernorm + clip
  ce_pool_ln<<<B_BATCH, 256, 0, stream>>>(partial, lengths, context);
}